// ConvEmbedding_2164663517774
// MI455X (gfx1250) — compile-verified
//
#include <hip/hip_runtime.h>

// ConvEmbedding: out[b,t,c] = bias[c] + sum_{k=0..8} W[c, x[b,t+k-4], k]
// x(8,4096) i32, W(512,32000,9) f32, b(512) f32 -> out(8,4096,512) f32.
//
// Memory-bound gather (~670MB @ 23.3TB/s ~= 29us); WMMA inapplicable
// (one-hot matmul == gather). Scatter/rolling-ring form: each token's 36B
// weight block W[c,id,0..8] is read once per channel as a 4B-aligned 36B
// memcpy (widest legal global loads, no rotation selects), accumulated into
// a 9-slot register ring. Token ids are block-uniform -> readfirstlane for
// scalar (saddr) addressing. NT stores for the write-once output; id-driven
// global_prefetch_b8 since gather addresses are data-dependent.

namespace {
constexpr int Bn  = 8;
constexpr int Tn  = 4096;
constexpr int Hn  = 512;
constexpr int Vn  = 32000;
constexpr int Kn  = 9;
constexpr int PAD = 4;
constexpr int TT  = 64;             // output t positions per block
constexpr int NTOK = TT + 2 * PAD;  // 72 tokens per block (multiple of 9)
constexpr int NTHREADS = Hn;        // one thread per output channel
constexpr int PFD = 18;             // prefetch distance in tokens
constexpr long long VK = (long long)Vn * Kn;  // 288000 floats per channel slice
} // namespace

__global__ __launch_bounds__(NTHREADS)
void conv_embed_kernel(const int* __restrict__ x,
                       const float* __restrict__ W,
                       const float* __restrict__ bias,
                       float* __restrict__ out) {
  __shared__ int s_off[NTOK];  // id*Kn: float offset of W[.,id,0] within a slice

  const int c  = threadIdx.x;  // channel; lanes contiguous in c -> coalesced stores
  const int b  = blockIdx.y;
  const int t0 = blockIdx.x * TT;

  // Stage the token window [t0-PAD, t0+TT+PAD). Clamp only so the x-load is
  // in bounds; out-of-range s contributes zero via the scalar branch below.
  for (int i = threadIdx.x; i < NTOK; i += NTHREADS) {
    int s  = t0 - PAD + i;
    int sc = s < 0 ? 0 : (s >= Tn ? Tn - 1 : s);
    s_off[i] = x[(size_t)b * Tn + sc] * Kn;
  }
  __syncthreads();

  const float  bc  = bias[c];
  const size_t cVK = (size_t)c * VK;  // per-lane slice offset (floats)
  const float* __restrict__ obase = out + ((size_t)b * Tn + t0) * Hn;

  // ring[(t - t0) % 9] accumulates output t: slot reset at i = t-t0 (fused
  // with its first add), retired (stored) at i = t-t0+8.
  float ring[9];
#pragma unroll
  for (int r = 0; r < 9; ++r) ring[r] = bc;

  for (int chunk = 0; chunk < NTOK / 9; ++chunk) {
#pragma unroll
    for (int u = 0; u < 9; ++u) {
      const int i = chunk * 9 + u;   // token iteration, i % 9 == u (static)
      const int s = t0 - PAD + i;    // token position (scalar-uniform)

      // Data-dependent software prefetch (clamped ids -> always a valid addr).
      const int ipf = i + PFD;
      if (ipf < NTOK) {
        const int opf = __builtin_amdgcn_readfirstlane(s_off[ipf]);
        __builtin_prefetch(W + opf + cVK, 0, 1);
      }

      if (s >= 0 && s < Tn) {        // scalar-uniform branch, no divergence
        // Uniform base (SGPR) + per-lane slice offset -> saddr+voffset loads.
        const int os = __builtin_amdgcn_readfirstlane(s_off[i]);
        const float* __restrict__ Wo = W + os + cVK;
        float wv[9];                 // W[c, id, 0..8]; 4B-aligned 36B block
        __builtin_memcpy(wv, Wo, sizeof(wv));
        // Token s feeds out[s+4-k] with tap k: slot (u+1+d)%9 gets wv[8-d].
        ring[u]           = bc + wv[0];  // fused reset + first add (t = t0+i)
        ring[(u + 1) % 9] += wv[8];
        ring[(u + 2) % 9] += wv[7];
        ring[(u + 3) % 9] += wv[6];
        ring[(u + 4) % 9] += wv[5];
        ring[(u + 5) % 9] += wv[4];
        ring[(u + 6) % 9] += wv[3];
        ring[(u + 7) % 9] += wv[2];
        ring[(u + 8) % 9] += wv[1];
      } else {
        ring[u] = bc;                // zero-padding: no contribution
      }

      // Retire output t = t0 + i - 8 (slot (i-8)%9 == (u+1)%9, static).
      if (i >= 8) {
        __builtin_nontemporal_store(ring[(u + 1) % 9],
                                    obase + (size_t)(i - 8) * Hn + c);
      }
    }
  }
}

extern "C" void kernel_launch(void* const* d_in, const int* in_sizes, int n_in,
                              void* d_out, int out_size, void* d_ws, size_t ws_size,
                              hipStream_t stream) {
  (void)in_sizes; (void)n_in; (void)out_size; (void)d_ws; (void)ws_size;
  const int*   x = (const int*)d_in[0];
  const float* W = (const float*)d_in[1];
  const float* b = (const float*)d_in[2];
  float* o = (float*)d_out;

  dim3 grid(Tn / TT, Bn);   // (64, 8) blocks of 512 threads (16 wave32)
  conv_embed_kernel<<<grid, NTHREADS, 0, stream>>>(x, W, b, o);
}